// LIFNode_39324720562288
// MI455X (gfx1250) — compile-verified
//
#include <hip/hip_runtime.h>
#include <stdint.h>

// LIF neuron forward scan on MI455X (gfx1250).
//
// x, out: [B=32, N=16384, T=100] float32, time contiguous.
// Memory-bound (AI ~= 0.5 flop/byte; ~420 MB traffic -> ~18us floor at
// 23.3 TB/s). Optimize purely for data movement:
//   Phase 1: async global->LDS DMA (coalesced 512B/wave-instr, ASYNCcnt path)
//   Phase 2: per-thread 100-step LIF scan in odd-pitch LDS rows (in-place,
//            bank-conflict-free)
//   Phase 3: coalesced non-temporal 128-bit stores

typedef float v4f __attribute__((ext_vector_type(4)));

#define ROWS_PER_BLOCK 128      // 4 wave32s
#define T_STEPS        100
#define ROW_PITCH      101      // dwords; odd pitch -> conflict-free scan
#define CHUNKS_PER_ROW 25       // 100 floats / 4 (16B chunks)
#define ITERS          25       // CHUNKS_PER_BLOCK / ROWS_PER_BLOCK
// per-iteration j stride 128 = 5*25 + 3  ->  row += 5, c += 3 (wrap at 25)
#define ROW_STEP       5
#define C_STEP         3
// LDS byte address step: 5 rows * 404 B + 3 chunks * 16 B
#define LDS_STEP       (ROW_STEP * ROW_PITCH * 4 + C_STEP * 16)   // 2068
// on c-wrap: row += 1 extra, c -= 25  ->  +404 - 400 = +4 bytes
#define LDS_WRAP_FIX   (ROW_PITCH * 4 - CHUNKS_PER_ROW * 16)      // 4

__global__ __launch_bounds__(ROWS_PER_BLOCK)
void lif_scan_kernel(const float* __restrict__ x, float* __restrict__ out) {
    __shared__ float tile[ROWS_PER_BLOCK * ROW_PITCH];   // 51,712 B

    const size_t block_base = (size_t)blockIdx.x * (ROWS_PER_BLOCK * T_STEPS);
    const float* __restrict__ xin  = x   + block_base;
    float*       __restrict__ outp = out + block_base;

    const int tid = threadIdx.x;
    // one-time div/mod for the starting chunk of this lane
    const int row0 = tid / CHUNKS_PER_ROW;
    const int c0   = tid - row0 * CHUNKS_PER_ROW;

    // LDS byte offset of tile[0] (generic LDS pointer: low 32 bits = offset)
    const uint32_t lds_base = (uint32_t)(size_t)(void*)&tile[0];

    // ---- Phase 1: async DMA of the block's contiguous 51,200B input region
    // into LDS, de-linearized into 101-dword padded rows. Global side is
    // perfectly coalesced (consecutive lanes -> consecutive 16B chunks).
    {
        uint32_t lds_addr = lds_base + (uint32_t)(row0 * ROW_PITCH + c0 * 4) * 4u;
        uint64_t gaddr    = (uint64_t)(uintptr_t)(xin + (size_t)tid * 4u);
        int c = c0;
#pragma unroll
        for (int i = 0; i < ITERS; ++i) {
            asm volatile("global_load_async_to_lds_b128 %0, %1, off"
                         :: "v"(lds_addr), "v"(gaddr)
                         : "memory");
            gaddr    += (uint64_t)ROWS_PER_BLOCK * 16u;   // +2048 B global
            lds_addr += LDS_STEP;
            c        += C_STEP;
            if (c >= CHUNKS_PER_ROW) { c -= CHUNKS_PER_ROW; lds_addr += LDS_WRAP_FIX; }
        }
    }
    // Wave-local DMA completion, then cross-wave visibility.
    asm volatile("s_wait_asynccnt 0x0" ::: "memory");
    __syncthreads();

    // ---- Phase 2: one thread per row, sequential LIF scan, spikes written
    // back in place (odd row pitch -> conflict-free LDS banking).
    {
        float* __restrict__ rp = &tile[tid * ROW_PITCH];
        float v = 0.0f;
#pragma unroll 4
        for (int k = 0; k < T_STEPS; ++k) {
            const float xv = rp[k];
            v = v * 0.5f + xv;                                // decay = 1 - 1/TAU
            const float s = (v - 0.5f) > 0.0f ? 1.0f : 0.0f;  // spike_fn(v - V_TH)
            v = v - s * 0.5f;                                 // soft reset
            rp[k] = s;
        }
    }
    __syncthreads();

    // ---- Phase 3: coalesced, non-temporal 128-bit stores of the spike tile
    // (write-once output; 420MB stream > 192MB L2, keep it low priority).
    {
        int row = row0, c = c0;
        float* gp = outp + (size_t)tid * 4u;
#pragma unroll
        for (int i = 0; i < ITERS; ++i) {
            const float* sp = &tile[row * ROW_PITCH + c * 4];
            v4f val;
            val.x = sp[0]; val.y = sp[1]; val.z = sp[2]; val.w = sp[3];
            __builtin_nontemporal_store(val, (v4f*)gp);
            gp  += ROWS_PER_BLOCK * 4;                  // +512 floats
            row += ROW_STEP;
            c   += C_STEP;
            if (c >= CHUNKS_PER_ROW) { c -= CHUNKS_PER_ROW; ++row; }
        }
    }
}

extern "C" void kernel_launch(void* const* d_in, const int* in_sizes, int n_in,
                              void* d_out, int out_size, void* d_ws, size_t ws_size,
                              hipStream_t stream) {
    (void)n_in; (void)d_ws; (void)ws_size; (void)out_size;
    const float* x = (const float*)d_in[0];
    float* out = (float*)d_out;

    const int total  = in_sizes[0];              // 32*16384*100 = 52,428,800
    const int rows   = total / T_STEPS;          // 524,288
    const int blocks = rows / ROWS_PER_BLOCK;    // 4,096

    lif_scan_kernel<<<blocks, ROWS_PER_BLOCK, 0, stream>>>(x, out);
}